// TensorDLT_44504451121194
// MI455X (gfx1250) — compile-verified
//
#include <hip/hip_runtime.h>
#include <hip/hip_bf16.h>
#include <stdint.h>

#define SYS_PER_BLOCK 256

// ---- CDNA5 async global<->LDS DMA helpers (ASYNCcnt-tracked) ----
__device__ __forceinline__ void async_load_lds_b128(unsigned lds_off, const void* gaddr) {
#if defined(__HIP_DEVICE_COMPILE__)
  asm volatile("global_load_async_to_lds_b128 %0, %1, off"
               :: "v"(lds_off), "v"(gaddr) : "memory");
#endif
}

__device__ __forceinline__ void async_store_lds_b32(const void* gaddr, unsigned lds_off) {
#if defined(__HIP_DEVICE_COMPILE__)
  asm volatile("global_store_async_from_lds_b32 %0, %1, off"
               :: "v"(gaddr), "v"(lds_off) : "memory");
#endif
}

__device__ __forceinline__ void wait_async0() {
#if defined(__HIP_DEVICE_COMPILE__)
  asm volatile("s_wait_asynccnt 0" ::: "memory");
#endif
}

// Closed-form 4-point homography (DLT) via null-space reduction:
//   P = [u v 1] (4x3).  System decouples into:
//     P*(h1,h2,h3) = up .* s ,  P*(h4,h5,h6) = vp .* s ,  s = 1 + u*h7 + v*h8
//   n = null(P^T) from signed 3x3 minors -> 2x2 solve for (h7,h8),
//   then two Cramer 3x3 solves sharing one determinant.
__global__ __launch_bounds__(SYS_PER_BLOCK) void
tensor_dlt_kernel(const float* __restrict__ corners,
                  const float* __restrict__ delta,
                  float* __restrict__ out, int nsys) {
  __shared__ float sIn[SYS_PER_BLOCK * 16];   // [0,2048): corners, [2048,4096): delta
  __shared__ float sOut[SYS_PER_BLOCK * 9];   // 2304 floats, mirrors global layout

  const int t = threadIdx.x;
  const int S = blockIdx.x * SYS_PER_BLOCK;
  const int nvalid = min(SYS_PER_BLOCK, nsys - S);

  const float* gC = corners + (size_t)S * 8;
  const float* gD = delta   + (size_t)S * 8;

  // Stage inputs: 512 b128 chunks per array per block, coalesced 512B/wave/instr.
#pragma unroll
  for (int k = 0; k < 2; ++k) {
    const int idx = k * SYS_PER_BLOCK + t;     // b128 chunk id (4 floats)
    if (idx < nvalid * 2) {
      async_load_lds_b128((unsigned)(size_t)&sIn[idx * 4],        gC + idx * 4);
      async_load_lds_b128((unsigned)(size_t)&sIn[2048 + idx * 4], gD + idx * 4);
    }
  }
  wait_async0();
  __syncthreads();

  if (t < nvalid) {
    const float* mc = &sIn[t * 8];
    const float* md = &sIn[2048 + t * 8];
    float u[4], v[4], up[4], vp[4];
#pragma unroll
    for (int j = 0; j < 4; ++j) {
      u[j]  = mc[2 * j];          v[j]  = mc[2 * j + 1];
      up[j] = u[j] + md[2 * j];   vp[j] = v[j] + md[2 * j + 1];
    }

    // Null vector n of P^T via signed minors (cofactor expansion).
    const float e23v = v[2] - v[3], e23u = u[2] - u[3], c23 = u[2] * v[3] - u[3] * v[2];
    const float e13v = v[1] - v[3], e13u = u[1] - u[3], c13 = u[1] * v[3] - u[3] * v[1];
    const float e12v = v[1] - v[2], e12u = u[1] - u[2], c12 = u[1] * v[2] - u[2] * v[1];
    const float n0 =  (u[1] * e23v - v[1] * e23u + c23);
    const float n1 = -(u[0] * e23v - v[0] * e23u + c23);
    const float n2 =  (u[0] * e13v - v[0] * e13u + c13);
    const float d3 =  (u[0] * e12v - v[0] * e12u + c12);  // det of rows 0,1,2 of P
    const float n3 = -d3;

    // 2x2 system for (h7,h8):  [n.*up | n.*vp] consistency conditions.
    const float a0 = n0 * up[0], a1 = n1 * up[1], a2 = n2 * up[2], a3 = n3 * up[3];
    const float b0 = n0 * vp[0], b1 = n1 * vp[1], b2 = n2 * vp[2], b3 = n3 * vp[3];
    const float A11 = a0 * u[0] + a1 * u[1] + a2 * u[2] + a3 * u[3];
    const float A12 = a0 * v[0] + a1 * v[1] + a2 * v[2] + a3 * v[3];
    const float A21 = b0 * u[0] + b1 * u[1] + b2 * u[2] + b3 * u[3];
    const float A22 = b0 * v[0] + b1 * v[1] + b2 * v[2] + b3 * v[3];
    const float r1 = -(a0 + a1 + a2 + a3);
    const float r2 = -(b0 + b1 + b2 + b3);
    const float inv2 = 1.0f / (A11 * A22 - A12 * A21);
    const float h7 = (r1 * A22 - A12 * r2) * inv2;
    const float h8 = (A11 * r2 - r1 * A21) * inv2;

    // Two 3x3 Cramer solves sharing det d3 (rows 0..2 of P).
    float ra[3], rb[3];
#pragma unroll
    for (int j = 0; j < 3; ++j) {
      const float s = 1.0f + u[j] * h7 + v[j] * h8;
      ra[j] = up[j] * s;
      rb[j] = vp[j] * s;
    }
    const float c02 = u[0] * v[2] - u[2] * v[0];
    const float c01 = u[0] * v[1] - u[1] * v[0];
    const float invd3 = 1.0f / d3;
    const float dra = ra[1] - ra[2], drb = rb[1] - rb[2];
    const float h1 = (ra[0] * e12v - v[0] * dra + (ra[1] * v[2] - ra[2] * v[1])) * invd3;
    const float h2 = (u[0] * dra - ra[0] * e12u + (u[1] * ra[2] - u[2] * ra[1])) * invd3;
    const float h3 = (ra[0] * c12 - ra[1] * c02 + ra[2] * c01) * invd3;
    const float h4 = (rb[0] * e12v - v[0] * drb + (rb[1] * v[2] - rb[2] * v[1])) * invd3;
    const float h5 = (u[0] * drb - rb[0] * e12u + (u[1] * rb[2] - u[2] * rb[1])) * invd3;
    const float h6 = (rb[0] * c12 - rb[1] * c02 + rb[2] * c01) * invd3;

    float* o = &sOut[t * 9];
    o[0] = h1; o[1] = h2; o[2] = h3;
    o[3] = h4; o[4] = h5; o[5] = h6;
    o[6] = h7; o[7] = h8; o[8] = 1.0f;
  }
  __syncthreads();

  // Drain results LDS -> HBM via async stores: coalesced 128B/wave/instr,
  // no VGPR round-trip despite the 9-float (36B) per-system stride.
  float* gO = out + (size_t)S * 9;
#pragma unroll
  for (int k = 0; k < 9; ++k) {
    const int idx = k * SYS_PER_BLOCK + t;
    if (idx < nvalid * 9)
      async_store_lds_b32(gO + idx, (unsigned)(size_t)&sOut[idx]);
  }
  wait_async0();
}

extern "C" void kernel_launch(void* const* d_in, const int* in_sizes, int n_in,
                              void* d_out, int out_size, void* d_ws, size_t ws_size,
                              hipStream_t stream) {
  const float* corners = (const float*)d_in[0];  // [B,4,2] f32
  const float* delta   = (const float*)d_in[1];  // [B,4,2] f32
  float* out = (float*)d_out;                    // [B,3,3] f32

  const int nsys = in_sizes[0] / 8;
  const int grid = (nsys + SYS_PER_BLOCK - 1) / SYS_PER_BLOCK;
  if (grid > 0) {
    tensor_dlt_kernel<<<grid, SYS_PER_BLOCK, 0, stream>>>(corners, delta, out, nsys);
  }
}